// TorchVQC_83897891160832
// MI455X (gfx1250) — compile-verified
//
#include <hip/hip_runtime.h>

typedef __attribute__((ext_vector_type(2))) float v2f;
typedef __attribute__((ext_vector_type(8))) float v8f;

#define WPB 8      // waves per block (wave32)
#define DD  256    // state dimension = 16x16

// ---------------- WMMA core: D = A(16x4) @ B(4x16) + C ----------------
static __device__ __forceinline__ v8f wmma4(v2f a, v2f b, v8f c) {
  return __builtin_amdgcn_wmma_f32_16x16x4_f32(false, a, false, b, (short)0, c,
                                               false, false);
}

// A fragment / B fragment loaders.
// Row-major 16x16 matrix at `base` (addr = row*16+col): element [i, 4k+2h+d]
static __device__ __forceinline__ v2f ldRow2(const float* base, int i, int k, int h) {
  const float* p = base + i * 16 + 4 * k + 2 * h;
  v2f r; r.x = p[0]; r.y = p[1];
  return r;
}
// Col-major 16x16 state at `base` (addr = col*16+row): A[m, kk] = base[kk*16+m]
static __device__ __forceinline__ v2f ldColS(const float* base, int m, int k, int h) {
  int kk = 4 * k + 2 * h;
  v2f r; r.x = base[kk * 16 + m]; r.y = base[(kk + 1) * 16 + m];
  return r;
}

// acc += A(16x16) @ B(16x16) via 4 chained 16x16x4 WMMAs.
template <bool NEGA, bool COLA>
static __device__ __forceinline__ v8f mm16(const float* A, const float* B, v8f acc,
                                           int i, int h) {
#pragma unroll
  for (int k = 0; k < 4; ++k) {
    v2f a = COLA ? ldColS(A, i, k, h) : ldRow2(A, i, k, h);
    if (NEGA) a = -a;
    v2f b = ldRow2(B, i, k, h);
    acc = wmma4(a, b, acc);
  }
  return acc;
}

// Store C/D fragment to col-major state: lane holds col i, rows 8h..8h+7.
static __device__ __forceinline__ void stCD(float* D, v8f v, int i, int h) {
  float* p = D + i * 16 + h * 8;
#pragma unroll
  for (int g = 0; g < 8; ++g) p[g] = v[g];
}

// D = L @ S (complex). L row-major in LDS, S col-major, D col-major.
static __device__ __forceinline__ void cmatL(const float* Lr, const float* Li,
                                             const float* Sr, const float* Si,
                                             float* Dr, float* Di, int lane) {
  int i = lane & 15, h = lane >> 4;
  v8f ar = {};
  ar = mm16<false, false>(Lr, Sr, ar, i, h);
  ar = mm16<true,  false>(Li, Si, ar, i, h);   // - Li*Si
  v8f ai = {};
  ai = mm16<false, false>(Lr, Si, ai, i, h);
  ai = mm16<false, false>(Li, Sr, ai, i, h);
  stCD(Dr, ar, i, h);
  stCD(Di, ai, i, h);
}

// D = T @ R^T (complex). T col-major, R row-major, D col-major.
static __device__ __forceinline__ void cmatR(const float* Tr, const float* Ti,
                                             const float* Rr, const float* Ri,
                                             float* Dr, float* Di, int lane) {
  int i = lane & 15, h = lane >> 4;
  v8f ar = {};
  ar = mm16<false, true>(Tr, Rr, ar, i, h);
  ar = mm16<true,  true>(Ti, Ri, ar, i, h);    // - Ti*Ri
  v8f ai = {};
  ai = mm16<false, true>(Tr, Ri, ai, i, h);
  ai = mm16<false, true>(Ti, Rr, ai, i, h);
  stCD(Dr, ar, i, h);
  stCD(Di, ai, i, h);
}

// ---------------- gate construction ----------------
// Rot = RZ(om) @ RY(th) @ RZ(phi)
static __device__ __forceinline__ void rotMat(float phi, float th, float om,
                                              float gr[2][2], float gi[2][2]) {
  float st, ct, sa, ca, sb, cb;
  __sincosf(th * 0.5f, &st, &ct);
  __sincosf(0.5f * (phi + om), &sa, &ca);
  __sincosf(0.5f * (phi - om), &sb, &cb);
  gr[0][0] = ct * ca;  gi[0][0] = -ct * sa;   //  c e^{-i(phi+om)/2}
  gr[0][1] = -st * cb; gi[0][1] = -st * sb;   // -s e^{+i(phi-om)/2}
  gr[1][0] = st * cb;  gi[1][0] = -st * sb;   //  s e^{-i(phi-om)/2}
  gr[1][1] = ct * ca;  gi[1][1] = ct * sa;    //  c e^{+i(phi+om)/2}
}

// G = RZ(phi) @ RY(th)  (encoding-style per-qubit gate)
static __device__ __forceinline__ void encMat(float th, float phi,
                                              float gr[2][2], float gi[2][2]) {
  float s, c, sp, cp;
  __sincosf(th * 0.5f, &s, &c);
  __sincosf(phi * 0.5f, &sp, &cp);
  gr[0][0] = c * cp;  gi[0][0] = -c * sp;
  gr[0][1] = -s * cp; gi[0][1] = s * sp;
  gr[1][0] = s * cp;  gi[1][0] = s * sp;
  gr[1][1] = c * cp;  gi[1][1] = c * sp;
}

// K = g0 (x) g1 (x) g2 (x) g3, stored row-major (addr = row*16+col) in LDS.
static __device__ __forceinline__ void buildKron(const float gr[4][2][2],
                                                 const float gi[4][2][2],
                                                 float* Kr, float* Ki, int lane) {
#pragma unroll
  for (int j = 0; j < 8; ++j) {
    int idx = lane * 8 + j;
    int a = idx >> 4;    // row
    int bcol = idx & 15; // col
    float pr = 1.f, pi = 0.f;
#pragma unroll
    for (int q = 0; q < 4; ++q) {
      int ba = (a >> (3 - q)) & 1;
      int bb = (bcol >> (3 - q)) & 1;
      float xr = gr[q][ba][bb], xi = gi[q][ba][bb];
      float nr = pr * xr - pi * xi;
      pi = pr * xi + pi * xr;
      pr = nr;
    }
    Kr[idx] = pr;
    Ki[idx] = pi;
  }
}

static __device__ __forceinline__ void buildRot(const float* vqcw, int l,
                                                float* GLr, float* GLi,
                                                float* GRr, float* GRi, int lane) {
  float gr[4][2][2], gi[4][2][2];
#pragma unroll
  for (int q = 0; q < 4; ++q) {
    const float* w = vqcw + (l * 8 + q) * 3;
    rotMat(w[0], w[1], w[2], gr[q], gi[q]);
  }
  buildKron(gr, gi, GLr, GLi, lane);
#pragma unroll
  for (int q = 0; q < 4; ++q) {
    const float* w = vqcw + (l * 8 + 4 + q) * 3;
    rotMat(w[0], w[1], w[2], gr[q], gi[q]);
  }
  buildKron(gr, gi, GRr, GRi, lane);
}

static __device__ __forceinline__ void buildReenc(const float* th, const float* ph,
                                                  float* GLr, float* GLi,
                                                  float* GRr, float* GRi, int lane) {
  float gr[4][2][2], gi[4][2][2];
#pragma unroll
  for (int q = 0; q < 4; ++q) encMat(th[q] * 0.5f, ph[q] * 0.5f, gr[q], gi[q]);
  buildKron(gr, gi, GLr, GLi, lane);
#pragma unroll
  for (int q = 0; q < 4; ++q) encMat(th[4 + q] * 0.5f, ph[4 + q] * 0.5f, gr[q], gi[q]);
  buildKron(gr, gi, GRr, GRi, lane);
}

// Composed CNOT ring permutation gather (canonical col-major LDS layout).
template <int RNG>
static __device__ __forceinline__ void applyCnot(const float* Sr, const float* Si,
                                                 float* Dr, float* Di, int lane) {
#pragma unroll
  for (int j = 0; j < 8; ++j) {
    int d = lane * 8 + j;
    int s = ((d & 15) << 4) | (d >> 4);  // addr -> state index
    int src = s;
#pragma unroll
    for (int k = 7; k >= 0; --k) {       // idx0[idx1[...idx7[i]]]
      int tgt = (k + RNG) & 7;
      src ^= ((src >> (7 - k)) & 1) << (7 - tgt);
    }
    int sa = ((src & 15) << 4) | (src >> 4);
    Dr[d] = Sr[sa];
    Di[d] = Si[sa];
  }
}

// ---------------- main kernel: one wave per batch element ----------------
extern "C" __global__ __launch_bounds__(256) void vqc_kernel(
    const float* __restrict__ theta, const float* __restrict__ phi,
    const float* __restrict__ jup, const float* __restrict__ vqcw,
    float* __restrict__ out) {
  __shared__ float sXr[WPB][DD], sXi[WPB][DD];
  __shared__ float sYr[WPB][DD], sYi[WPB][DD];
  __shared__ float sGLr[WPB][DD], sGLi[WPB][DD];
  __shared__ float sGRr[WPB][DD], sGRi[WPB][DD];

  int tid = threadIdx.x;
  int wid = tid >> 5, lane = tid & 31;
  int b = blockIdx.x * WPB + wid;

  const float* th = theta + b * 8;
  const float* ph = phi + b * 8;
  const float* jj = jup + b * 28;

  float* Xr = sXr[wid]; float* Xi = sXi[wid];
  float* Yr = sYr[wid]; float* Yi = sYi[wid];
  float* GLr = sGLr[wid]; float* GLi = sGLi[wid];
  float* GRr = sGRr[wid]; float* GRi = sGRi[wid];

  // ---- phase 0: encoding product state * IsingZZ phase -> X ----
  {
    float e0r[8], e0i[8], e1r[8], e1i[8];
#pragma unroll
    for (int k = 0; k < 8; ++k) {
      float s, c, sp, cp;
      __sincosf(th[k] * 0.5f, &s, &c);
      __sincosf(ph[k] * 0.5f, &sp, &cp);
      e0r[k] = c * cp; e0i[k] = -c * sp;   // G[0][0]
      e1r[k] = s * cp; e1i[k] = s * sp;    // G[1][0]
    }
    float jv[28];
#pragma unroll
    for (int p = 0; p < 28; ++p) jv[p] = jj[p];

#pragma unroll
    for (int j = 0; j < 8; ++j) {
      int d = lane * 8 + j;
      int s = ((d & 15) << 4) | (d >> 4);
      float pr = 1.f, pim = 0.f;
#pragma unroll
      for (int k = 0; k < 8; ++k) {
        int bit = (s >> (7 - k)) & 1;
        float xr = bit ? e1r[k] : e0r[k];
        float xi = bit ? e1i[k] : e0i[k];
        float nr = pr * xr - pim * xi;
        pim = pr * xi + pim * xr;
        pr = nr;
      }
      float e = 0.f;
      int p = 0;
#pragma unroll
      for (int qi = 0; qi < 8; ++qi)
#pragma unroll
        for (int qj = qi + 1; qj < 8; ++qj) {
          int diff = ((s >> (7 - qi)) & 1) ^ ((s >> (7 - qj)) & 1);
          e += diff ? jv[p] : -jv[p];
          ++p;
        }
      e *= 1.57079632679489662f;  // pi/2
      float se, ce;
      __sincosf(e, &se, &ce);
      Xr[d] = pr * ce - pim * se;
      Xi[d] = pr * se + pim * ce;
    }
  }
  buildRot(vqcw, 0, GLr, GLi, GRr, GRi, lane);
  __syncthreads();

  // ---- layer 0: Rot ----
  cmatL(GLr, GLi, Xr, Xi, Yr, Yi, lane);
  __syncthreads();
  cmatR(Yr, Yi, GRr, GRi, Xr, Xi, lane);
  __syncthreads();

  // ---- layer 0: CNOT ring (rng=1), build re-encoding gates ----
  applyCnot<1>(Xr, Xi, Yr, Yi, lane);
  buildReenc(th, ph, GLr, GLi, GRr, GRi, lane);
  __syncthreads();

  // ---- re-encoding ----
  cmatL(GLr, GLi, Yr, Yi, Xr, Xi, lane);
  __syncthreads();
  cmatR(Xr, Xi, GRr, GRi, Yr, Yi, lane);
  __syncthreads();

  // ---- layer 1: Rot ----
  buildRot(vqcw, 1, GLr, GLi, GRr, GRi, lane);
  __syncthreads();
  cmatL(GLr, GLi, Yr, Yi, Xr, Xi, lane);
  __syncthreads();
  cmatR(Xr, Xi, GRr, GRi, Yr, Yi, lane);
  __syncthreads();

  // ---- layer 1: CNOT ring (rng=2) ----
  applyCnot<2>(Yr, Yi, Xr, Xi, lane);
  __syncthreads();

  // ---- readout: probs @ Z_SIGNS^T, wave reduction ----
  float acc[8];
#pragma unroll
  for (int k = 0; k < 8; ++k) acc[k] = 0.f;
#pragma unroll
  for (int j = 0; j < 8; ++j) {
    int d = lane * 8 + j;
    int s = ((d & 15) << 4) | (d >> 4);
    float vr = Xr[d], vi = Xi[d];
    float p2 = vr * vr + vi * vi;
#pragma unroll
    for (int k = 0; k < 8; ++k)
      acc[k] += ((s >> (7 - k)) & 1) ? -p2 : p2;
  }
#pragma unroll
  for (int k = 0; k < 8; ++k) {
    float v = acc[k];
    v += __shfl_xor(v, 16, 32);
    v += __shfl_xor(v, 8, 32);
    v += __shfl_xor(v, 4, 32);
    v += __shfl_xor(v, 2, 32);
    v += __shfl_xor(v, 1, 32);
    acc[k] = v;
  }
  if (lane == 0) {
#pragma unroll
    for (int k = 0; k < 8; ++k) out[b * 8 + k] = acc[k];
  }
}

extern "C" void kernel_launch(void* const* d_in, const int* in_sizes, int n_in,
                              void* d_out, int out_size, void* d_ws, size_t ws_size,
                              hipStream_t stream) {
  const float* theta = (const float*)d_in[0];
  const float* phi   = (const float*)d_in[1];
  const float* jup   = (const float*)d_in[2];
  const float* vqcw  = (const float*)d_in[3];
  float* out = (float*)d_out;
  int B = in_sizes[0] / 8;          // 16384 batch elements
  dim3 grid(B / WPB), block(32 * WPB);
  hipLaunchKernelGGL(vqc_kernel, grid, block, 0, stream, theta, phi, jup, vqcw, out);
}